// DecorrLoss_1666447311133
// MI455X (gfx1250) — compile-verified
//
#include <hip/hip_runtime.h>
#include <hip/hip_bf16.h>

typedef __attribute__((ext_vector_type(2))) float v2f;
typedef __attribute__((ext_vector_type(4))) float v4f;
typedef __attribute__((ext_vector_type(8))) float v8f;

// ---------------------------------------------------------------------------
// Kernel 1: split-K partial Gram matrix G_c = sum_{k in chunk c} x_k x_k^T
// via V_WMMA_F32_16X16X4_F32 (fp32 WMMA == f32 reference math).
//
// Each wave owns a 32x32 output block = 2x2 WMMA tiles, so every K-step
// loads 2 A fragments + 2 B fragments (8 b32 loads/lane) and issues 4 WMMAs.
//   A (16x4 f32): lane&15 = M, lane>>4 picks K-half, 2 VGPRs = K in half
//   B (4x16 f32): lane&15 = N, same K mapping (symmetric layout)
//   C/D (16x16 f32): VGPR r, lane L -> M = r + 8*(L>>4), N = L&15
// grid = (2, nchunk); 8 waves/block -> wave-job = blockIdx.x*8 + wave (0..15)
// ---------------------------------------------------------------------------
__global__ __launch_bounds__(256) void decorr_gram_partial(
    const float* __restrict__ x,   // (n, 128) row-major
    float* __restrict__ ws_gram,   // (nchunk, 128, 128)
    int chunk_len)                 // samples per chunk (multiple of 4)
{
    const int lane = threadIdx.x & 31;
    // Force wave id into an SGPR so loop bounds are provably uniform
    // (scalar s_cbranch loop instead of an exec-masked vector loop).
    const int wave = __builtin_amdgcn_readfirstlane((int)(threadIdx.x >> 5));
    const int job  = blockIdx.x * 8 + wave;     // 0..15
    const int ti2  = job >> 2;                  // 32-row block (0..3)
    const int tj2  = job & 3;                   // 32-col block (0..3)

    const int m16   = lane & 15;
    const int khalf = (lane >> 4) << 1;         // 0 or 2

    const float* __restrict__ xA0 = x + (ti2 * 32 + m16);
    const float* __restrict__ xA1 = xA0 + 16;
    const float* __restrict__ xB0 = x + (tj2 * 32 + m16);
    const float* __restrict__ xB1 = xB0 + 16;

    v8f acc00 = {0.f,0.f,0.f,0.f,0.f,0.f,0.f,0.f};
    v8f acc01 = {0.f,0.f,0.f,0.f,0.f,0.f,0.f,0.f};
    v8f acc10 = {0.f,0.f,0.f,0.f,0.f,0.f,0.f,0.f};
    v8f acc11 = {0.f,0.f,0.f,0.f,0.f,0.f,0.f,0.f};

    const int kbeg = blockIdx.y * chunk_len;    // uniform (SGPR)
    const int kend = kbeg + chunk_len;

    #pragma unroll 2
    for (int k0 = kbeg; k0 < kend; k0 += 4) {
        const size_t r0 = (size_t)(k0 + khalf) * 128;
        const size_t r1 = r0 + 128;
        const v2f a0 = {xA0[r0], xA0[r1]};
        const v2f a1 = {xA1[r0], xA1[r1]};
        const v2f b0 = {xB0[r0], xB0[r1]};
        const v2f b1 = {xB1[r0], xB1[r1]};
        acc00 = __builtin_amdgcn_wmma_f32_16x16x4_f32(false, a0, false, b0,
                                                      (short)0, acc00, false, false);
        acc01 = __builtin_amdgcn_wmma_f32_16x16x4_f32(false, a0, false, b1,
                                                      (short)0, acc01, false, false);
        acc10 = __builtin_amdgcn_wmma_f32_16x16x4_f32(false, a1, false, b0,
                                                      (short)0, acc10, false, false);
        acc11 = __builtin_amdgcn_wmma_f32_16x16x4_f32(false, a1, false, b1,
                                                      (short)0, acc11, false, false);
    }

    float* __restrict__ outp = ws_gram + (size_t)blockIdx.y * (128 * 128);
    const int rbase = ti2 * 32 + ((lane >> 4) << 3);   // row of acc element 0
    const int cbase = tj2 * 32 + m16;
    #pragma unroll
    for (int r = 0; r < 8; ++r) {
        outp[(size_t)(rbase + r)      * 128 + cbase]      = acc00[r];
        outp[(size_t)(rbase + r)      * 128 + cbase + 16] = acc01[r];
        outp[(size_t)(rbase + r + 16) * 128 + cbase]      = acc10[r];
        outp[(size_t)(rbase + r + 16) * 128 + cbase + 16] = acc11[r];
    }
}

// ---------------------------------------------------------------------------
// Kernel 2: per-block partial loss sums. One b128 load per lane covers a full
// 128-float sample row per wave; shfl_xor wave reduction for s2/s4.
//   corr partial: s2^2 - s4     whit partial: s4 - 2*s2
// ---------------------------------------------------------------------------
__global__ __launch_bounds__(256) void decorr_loss_partial(
    const float* __restrict__ x,       // (n, 128)
    float* __restrict__ ws_loss,       // (gridDim.x, 2)
    int n)
{
    __shared__ float sc[8];
    __shared__ float sw[8];

    const int lane  = threadIdx.x & 31;
    const int wave  = threadIdx.x >> 5;
    const int wpb   = blockDim.x >> 5;
    const int gwave = blockIdx.x * wpb + wave;
    const int nwav  = gridDim.x * wpb;

    float corr_acc = 0.f;
    float whit_acc = 0.f;

    for (int s = gwave; s < n; s += nwav) {
        const v4f v = *(const v4f*)(x + (size_t)s * 128 + lane * 4);
        float s2p = 0.f, s4p = 0.f;
        #pragma unroll
        for (int j = 0; j < 4; ++j) {
            const float v2 = v[j] * v[j];
            s2p += v2;
            s4p += v2 * v2;
        }
        #pragma unroll
        for (int off = 16; off > 0; off >>= 1) {
            s2p += __shfl_xor(s2p, off, 32);
            s4p += __shfl_xor(s4p, off, 32);
        }
        corr_acc += s2p * s2p - s4p;
        whit_acc += s4p - 2.0f * s2p;
    }

    if (lane == 0) { sc[wave] = corr_acc; sw[wave] = whit_acc; }
    __syncthreads();
    if (threadIdx.x == 0) {
        float c = 0.f, w = 0.f;
        for (int i = 0; i < wpb; ++i) { c += sc[i]; w += sw[i]; }
        ws_loss[2 * blockIdx.x + 0] = c;
        ws_loss[2 * blockIdx.x + 1] = w;
    }
}

// ---------------------------------------------------------------------------
// Kernel 3: deterministic reduction + finalize.
//   grad[i][j] = (i==j) ? kappa*(G_ii/n - 1) : (1-kappa)*G_ij/n
//   out[16384] = corr, out[16385] = whit
// ---------------------------------------------------------------------------
__global__ __launch_bounds__(256) void decorr_finalize(
    const float* __restrict__ ws_gram,  // (nchunk, 128, 128)
    int nchunk,
    const float* __restrict__ ws_loss,  // (nblk, 2)
    int nblk,
    const float* __restrict__ kappa_ptr,
    float* __restrict__ out,            // 128*128 + 2
    int n)
{
    const int idx = blockIdx.x * blockDim.x + threadIdx.x;
    const float inv_n = 1.0f / (float)n;

    if (idx < 128 * 128) {
        float s = 0.f;
        for (int c = 0; c < nchunk; ++c)
            s += ws_gram[(size_t)c * (128 * 128) + idx];
        const float g = s * inv_n;
        const int i = idx >> 7;
        const int j = idx & 127;
        const float kappa = *kappa_ptr;
        out[idx] = (i == j) ? kappa * (g - 1.0f) : (1.0f - kappa) * g;
    }

    if (blockIdx.x == 0 && threadIdx.x == 0) {
        float cs = 0.f, wsum = 0.f;
        for (int b = 0; b < nblk; ++b) {
            cs   += ws_loss[2 * b + 0];
            wsum += ws_loss[2 * b + 1];
        }
        const float inv_nd2 = inv_n / (128.0f * 128.0f);
        out[128 * 128 + 0] = cs * inv_nd2;
        out[128 * 128 + 1] = (wsum + (float)n * 128.0f) * inv_nd2;
    }
}

extern "C" void kernel_launch(void* const* d_in, const int* in_sizes, int n_in,
                              void* d_out, int out_size, void* d_ws, size_t ws_size,
                              hipStream_t stream) {
    (void)n_in; (void)out_size;
    const float* x     = (const float*)d_in[0];
    const float* kappa = (const float*)d_in[1];
    float*       out   = (float*)d_out;

    const int d = 128;
    const int n = in_sizes[0] / d;          // 8192 samples

    const int NBLK_LOSS = 64;
    const size_t loss_bytes = (size_t)NBLK_LOSS * 2 * sizeof(float);
    const size_t gram_bytes = (size_t)d * d * sizeof(float);

    // Split-K factor: as many chunks as workspace allows (cap 32), chunk_len
    // kept a multiple of 4 for the 16x16x4 WMMA K-step.
    int nchunk = 32;
    while (nchunk > 1 &&
           ((size_t)nchunk * gram_bytes + loss_bytes > ws_size ||
            (n / nchunk) % 4 != 0)) {
        nchunk >>= 1;
    }
    const int chunk_len = n / nchunk;

    float* ws_gram = (float*)d_ws;
    float* ws_loss = (float*)((char*)d_ws + (size_t)nchunk * gram_bytes);

    // 16 wave-jobs (4x4 grid of 32x32 blocks) per chunk; 8 waves per block.
    dim3 gram_grid(2, nchunk);
    decorr_gram_partial<<<gram_grid, 256, 0, stream>>>(x, ws_gram, chunk_len);

    decorr_loss_partial<<<NBLK_LOSS, 256, 0, stream>>>(x, ws_loss, n);

    decorr_finalize<<<(d * d + 255) / 256, 256, 0, stream>>>(
        ws_gram, nchunk, ws_loss, NBLK_LOSS, kappa, out, n);
}